// Net_25769803776036
// MI455X (gfx1250) — compile-verified
//
#include <hip/hip_runtime.h>
#include <hip/hip_bf16.h>

#define ANNOT 512
#define HID 64
#define N_STEPS 8
#define N_NODES 100000
#define N_EDGES 1600000
#define NUM_GRAPHS 64
#define ROW_TILES (N_NODES / 16)          // 6250 exact 16-row tiles
#define GEMM_BLOCKS ((ROW_TILES + 3) / 4) // 4 tiles (waves) per block
#define GRU_TPB 8                          // row tiles per GRU block
#define GRU_BLOCKS ((ROW_TILES + GRU_TPB - 1) / GRU_TPB)
#define WPAD 68                            // 64 + 4 pad: conflict-free b64 LDS reads

typedef __attribute__((ext_vector_type(2))) float v2f;
typedef __attribute__((ext_vector_type(8))) float v8f;

__device__ __forceinline__ v8f wmma4(v2f a, v2f b, v8f c) {
    // V_WMMA_F32_16X16X4_F32 : D = A(16x4) * B(4x16) + C(16x16), fp32
    return __builtin_amdgcn_wmma_f32_16x16x4_f32(
        false, a, false, b, (short)0, c, false, false);
}

__device__ __forceinline__ float sigmoidf_(float x) {
    return 1.0f / (1.0f + expf(-x));
}

// ---------------------------------------------------------------------------
// C[N,64] = A[N,K] @ B[K,64] (+ bias).
// One wave per 16-row tile covering all 64 output cols (4 acc chains, shared
// A fragment).  B staged in LDS transposed+padded in 128-K chunks:
// fragment read = one conflict-free ds_load_b64 per col-tile.
// ---------------------------------------------------------------------------
template<int K>
__global__ void gemm_n64(const float* __restrict__ A, const float* __restrict__ B,
                         const float* __restrict__ bias, float* __restrict__ C) {
    constexpr int KC = (K > 128) ? 128 : K;
    constexpr int BSTR = KC + 4;
    __shared__ __align__(16) float Bs[64 * BSTR];

    const int t = threadIdx.x;
    const int wave = t >> 5, lane = t & 31;
    const int hl = lane >> 4, idx = lane & 15;
    const int tile = blockIdx.x * 4 + wave;
    const bool valid = (tile < ROW_TILES);   // uniform per wave -> EXEC stays all-1s

    v8f acc0 = {}, acc1 = {}, acc2 = {}, acc3 = {};
    const float* __restrict__ arow =
        A + (size_t)((valid ? tile : 0) * 16 + idx) * (size_t)K;

    for (int kb = 0; kb < K; kb += KC) {
        __syncthreads();
        // stage B[kb:kb+KC, 0:64] transposed: Bs[n*BSTR + kk] = B[(kb+kk)*64 + n]
        for (int i = t; i < 64 * KC; i += 128) {
            const int n = i & 63;
            const int kk = i >> 6;
            Bs[n * BSTR + kk] = B[(size_t)(kb + kk) * 64 + n];
        }
        __syncthreads();
        if (valid) {
            #pragma unroll 4
            for (int k0 = 0; k0 < KC; k0 += 4) {
                const int ka = k0 + 2 * hl;
                v2f a  = *(const v2f*)(arow + kb + ka);
                v2f b0 = *(const v2f*)&Bs[(0 * 16 + idx) * BSTR + ka];
                v2f b1 = *(const v2f*)&Bs[(1 * 16 + idx) * BSTR + ka];
                v2f b2 = *(const v2f*)&Bs[(2 * 16 + idx) * BSTR + ka];
                v2f b3 = *(const v2f*)&Bs[(3 * 16 + idx) * BSTR + ka];
                acc0 = wmma4(a, b0, acc0);
                acc1 = wmma4(a, b1, acc1);
                acc2 = wmma4(a, b2, acc2);
                acc3 = wmma4(a, b3, acc3);
            }
        }
    }

    if (valid) {
        const float bv0 = bias ? bias[ 0 + idx] : 0.0f;
        const float bv1 = bias ? bias[16 + idx] : 0.0f;
        const float bv2 = bias ? bias[32 + idx] : 0.0f;
        const float bv3 = bias ? bias[48 + idx] : 0.0f;
        const int row = tile * 16 + 8 * hl;
        #pragma unroll
        for (int r = 0; r < 8; ++r) {
            float* __restrict__ crow = C + (size_t)(row + r) * 64;
            crow[ 0 + idx] = acc0[r] + bv0;
            crow[16 + idx] = acc1[r] + bv1;
            crow[32 + idx] = acc2[r] + bv2;
            crow[48 + idx] = acc3[r] + bv3;
        }
    }
}

// ---------------------------------------------------------------------------
// aggr[dst] += m[src] over edges.  16 lanes per edge, float4 per lane.
// ---------------------------------------------------------------------------
__global__ void scatter_add_edges(const float* __restrict__ m,
                                  const int* __restrict__ src,
                                  const int* __restrict__ dst,
                                  float* __restrict__ aggr) {
    const long long tid = (long long)blockIdx.x * blockDim.x + threadIdx.x;
    const long long e = tid >> 4;
    if (e >= N_EDGES) return;
    const int c = (int)(tid & 15) * 4;
    const int s = src[e];
    const int d = dst[e];
    const float4 v = *(const float4*)(m + (size_t)s * 64 + c);
    float* out = aggr + (size_t)d * 64 + c;
    atomicAdd(out + 0, v.x);
    atomicAdd(out + 1, v.y);
    atomicAdd(out + 2, v.z);
    atomicAdd(out + 3, v.w);
}

__global__ void zero_f32(float* __restrict__ p, int n) {
    int i = blockIdx.x * blockDim.x + threadIdx.x;
    if (i < n) p[i] = 0.0f;
}

// ---------------------------------------------------------------------------
// Fused GRU cell: h_out = GRU(aggr, h).
// Block stages w_ih/w_hh once in LDS (padded, conflict-free b64 fragment
// reads), then loops GRU_TPB row tiles, staging aggr/h rows per tile.
// Wave = one 16-col chunk, six independent WMMA accumulator chains.
// Dynamic LDS: 2*192*WPAD + 2*16*WPAD floats.
// ---------------------------------------------------------------------------
__global__ void gru_fused(const float* __restrict__ aggr, const float* __restrict__ h,
                          const float* __restrict__ w_ih, const float* __restrict__ w_hh,
                          const float* __restrict__ b_ih, const float* __restrict__ b_hh,
                          float* __restrict__ h_out) {
    extern __shared__ __align__(16) float smem[];
    float* Wih = smem;                   // [192][WPAD]
    float* Whh = Wih + 192 * WPAD;       // [192][WPAD]
    float* As  = Whh + 192 * WPAD;       // [16][WPAD]
    float* Hs  = As  + 16 * WPAD;        // [16][WPAD]

    const int t = threadIdx.x;
    // stage weights (coalesced in k)
    for (int i = t; i < 192 * 64; i += 128) {
        const int n = i >> 6, k = i & 63;
        Wih[n * WPAD + k] = w_ih[i];
        Whh[n * WPAD + k] = w_hh[i];
    }

    const int wave = t >> 5, lane = t & 31;
    const int hl = lane >> 4, idx = lane & 15;
    const int c0 = wave * 16;

    const float bir_ = b_ih[0 * HID + c0 + idx];
    const float biz_ = b_ih[1 * HID + c0 + idx];
    const float bin_ = b_ih[2 * HID + c0 + idx];
    const float bhr_ = b_hh[0 * HID + c0 + idx];
    const float bhz_ = b_hh[1 * HID + c0 + idx];
    const float bhn_ = b_hh[2 * HID + c0 + idx];

    const size_t wr = (size_t)(0 * HID + c0 + idx) * WPAD;
    const size_t wz = (size_t)(1 * HID + c0 + idx) * WPAD;
    const size_t wn = (size_t)(2 * HID + c0 + idx) * WPAD;

    int tend = blockIdx.x * GRU_TPB + GRU_TPB;
    if (tend > ROW_TILES) tend = ROW_TILES;

    for (int tile = blockIdx.x * GRU_TPB; tile < tend; ++tile) {
        __syncthreads();   // also covers weight staging before first tile
        for (int i = t; i < 16 * 64; i += 128) {
            const int lr = i >> 6, k = i & 63;
            As[lr * WPAD + k] = aggr[(size_t)tile * 1024 + i];
            Hs[lr * WPAD + k] = h[(size_t)tile * 1024 + i];
        }
        __syncthreads();

        v8f ir = {}, iz = {}, inn = {}, hr = {}, hz = {}, hn = {};
        #pragma unroll 4
        for (int k0 = 0; k0 < HID; k0 += 4) {
            const int ka = k0 + 2 * hl;
            v2f aa  = *(const v2f*)&As[idx * WPAD + ka];
            v2f ah  = *(const v2f*)&Hs[idx * WPAD + ka];
            v2f bir = *(const v2f*)&Wih[wr + ka];
            v2f biz = *(const v2f*)&Wih[wz + ka];
            v2f bin = *(const v2f*)&Wih[wn + ka];
            v2f bhr = *(const v2f*)&Whh[wr + ka];
            v2f bhz = *(const v2f*)&Whh[wz + ka];
            v2f bhn = *(const v2f*)&Whh[wn + ka];
            ir  = wmma4(aa, bir, ir);
            iz  = wmma4(aa, biz, iz);
            inn = wmma4(aa, bin, inn);
            hr  = wmma4(ah, bhr, hr);
            hz  = wmma4(ah, bhz, hz);
            hn  = wmma4(ah, bhn, hn);
        }

        const int row = tile * 16 + 8 * hl;
        #pragma unroll
        for (int r = 0; r < 8; ++r) {
            const float rg = sigmoidf_(ir[r] + bir_ + hr[r] + bhr_);
            const float zg = sigmoidf_(iz[r] + biz_ + hz[r] + bhz_);
            const float ng = tanhf(inn[r] + bin_ + rg * (hn[r] + bhn_));
            const float hp = Hs[(r + 8 * hl) * WPAD + c0 + idx];
            h_out[(size_t)(row + r) * 64 + c0 + idx] = (1.0f - zg) * ng + zg * hp;
        }
    }
}

// ---------------------------------------------------------------------------
// Pooling stage kernels
// ---------------------------------------------------------------------------
__device__ __forceinline__ unsigned f32_order_key(float x) {
    unsigned u = __float_as_uint(x);
    return (u & 0x80000000u) ? ~u : (u | 0x80000000u);
}
__device__ __forceinline__ float f32_order_unkey(unsigned u) {
    return (u & 0x80000000u) ? __uint_as_float(u & 0x7FFFFFFFu) : __uint_as_float(~u);
}

__global__ void node_gate_feat(const float* __restrict__ h,
                               const float* __restrict__ gate_w, const float* __restrict__ gate_b,
                               const float* __restrict__ out_w,  const float* __restrict__ out_b,
                               const int* __restrict__ batch,
                               float* __restrict__ logit, float* __restrict__ feat,
                               unsigned* __restrict__ gmax) {
    const int i = blockIdx.x * blockDim.x + threadIdx.x;
    if (i >= N_NODES) return;
    const float4* __restrict__ h4 = (const float4*)(h + (size_t)i * 64);
    float l = gate_b[0], f0 = out_b[0], f1 = out_b[1];
    #pragma unroll
    for (int q = 0; q < 16; ++q) {
        const float4 v = h4[q];
        const int k = q * 4;
        l  += v.x * gate_w[k]     + v.y * gate_w[k + 1]
            + v.z * gate_w[k + 2] + v.w * gate_w[k + 3];
        f0 += v.x * out_w[2*k]     + v.y * out_w[2*k + 2]
            + v.z * out_w[2*k + 4] + v.w * out_w[2*k + 6];
        f1 += v.x * out_w[2*k + 1] + v.y * out_w[2*k + 3]
            + v.z * out_w[2*k + 5] + v.w * out_w[2*k + 7];
    }
    logit[i] = l;
    feat[2 * i + 0] = f0;
    feat[2 * i + 1] = f1;
    atomicMax(&gmax[batch[i]], f32_order_key(l));
}

__global__ void node_exp_sum(const float* __restrict__ logit, const int* __restrict__ batch,
                             const unsigned* __restrict__ gmax,
                             float* __restrict__ evals, float* __restrict__ ssum) {
    const int i = blockIdx.x * blockDim.x + threadIdx.x;
    if (i >= N_NODES) return;
    const int b = batch[i];
    const float mx = f32_order_unkey(gmax[b]);
    const float ev = expf(logit[i] - mx);
    evals[i] = ev;
    atomicAdd(&ssum[b], ev);
}

__global__ void node_pool(const float* __restrict__ evals, const float* __restrict__ feat,
                          const int* __restrict__ batch, const float* __restrict__ ssum,
                          float* __restrict__ pooled) {
    const int i = blockIdx.x * blockDim.x + threadIdx.x;
    if (i >= N_NODES) return;
    const int b = batch[i];
    const float g = evals[i] / (ssum[b] + 1e-16f);
    atomicAdd(&pooled[2 * b + 0], g * feat[2 * i + 0]);
    atomicAdd(&pooled[2 * b + 1], g * feat[2 * i + 1]);
}

__global__ void final_softmax(const float* __restrict__ pooled, float* __restrict__ out) {
    const int g = threadIdx.x;
    if (g >= NUM_GRAPHS) return;
    const float a = pooled[2 * g], b = pooled[2 * g + 1];
    const float mx = fmaxf(a, b);
    const float ea = expf(a - mx), eb = expf(b - mx);
    const float s = ea + eb;
    out[2 * g + 0] = ea / s;
    out[2 * g + 1] = eb / s;
}

// ---------------------------------------------------------------------------
// Launch
// ---------------------------------------------------------------------------
extern "C" void kernel_launch(void* const* d_in, const int* in_sizes, int n_in,
                              void* d_out, int out_size, void* d_ws, size_t ws_size,
                              hipStream_t stream) {
    const float* x        = (const float*)d_in[0];
    const int*   edge_idx = (const int*)d_in[1];     // [2, E]
    const int*   batch    = (const int*)d_in[2];     // [N]
    const float* reduce_w = (const float*)d_in[3];
    const float* reduce_b = (const float*)d_in[4];
    const float* ggc_w    = (const float*)d_in[5];   // [8,64,64]
    const float* gru_w_ih = (const float*)d_in[6];
    const float* gru_w_hh = (const float*)d_in[7];
    const float* gru_b_ih = (const float*)d_in[8];
    const float* gru_b_hh = (const float*)d_in[9];
    const float* gate_w   = (const float*)d_in[10];
    const float* gate_b   = (const float*)d_in[11];
    const float* out_w    = (const float*)d_in[12];
    const float* out_b    = (const float*)d_in[13];
    float* out = (float*)d_out;

    const int* src = edge_idx;
    const int* dst = edge_idx + N_EDGES;

    // Workspace layout (floats): ~79 MB total
    float* h_a   = (float*)d_ws;                 // [N,64]
    float* h_b   = h_a + (size_t)N_NODES * 64;   // [N,64]  (m / h_next ping-pong)
    float* aggr  = h_b + (size_t)N_NODES * 64;   // [N,64]
    float* logit = aggr + (size_t)N_NODES * 64;  // [N]
    float* evals = logit + N_NODES;              // [N]
    float* feat  = evals + N_NODES;              // [N,2]
    float* ssum  = feat + (size_t)2 * N_NODES;   // [G]
    float* pooled = ssum + NUM_GRAPHS;           // [G,2]
    unsigned* gmax = (unsigned*)(pooled + 2 * NUM_GRAPHS); // [G]

    const size_t gru_lds = (size_t)(2 * 192 * WPAD + 2 * 16 * WPAD) * sizeof(float);

    // 1) h = x @ reduce_w + reduce_b
    gemm_n64<ANNOT><<<GEMM_BLOCKS, 128, 0, stream>>>(x, reduce_w, reduce_b, h_a);

    // 2) 8 GatedGraphConv steps (ping-pong h_a <-> h_b)
    float* h_cur = h_a;
    float* h_nxt = h_b;
    const int edge_blocks = (N_EDGES * 16 + 255) / 256;
    const int nfeat = N_NODES * 64;
    for (int s = 0; s < N_STEPS; ++s) {
        gemm_n64<HID><<<GEMM_BLOCKS, 128, 0, stream>>>(h_cur, ggc_w + (size_t)s * HID * HID,
                                                       nullptr, h_nxt);
        zero_f32<<<(nfeat + 255) / 256, 256, 0, stream>>>(aggr, nfeat);
        scatter_add_edges<<<edge_blocks, 256, 0, stream>>>(h_nxt, src, dst, aggr);
        gru_fused<<<GRU_BLOCKS, 128, gru_lds, stream>>>(aggr, h_cur, gru_w_ih, gru_w_hh,
                                                        gru_b_ih, gru_b_hh, h_nxt);
        float* tp = h_cur; h_cur = h_nxt; h_nxt = tp;
    }

    // 3) GlobalAttention pooling
    hipMemsetAsync(gmax, 0, NUM_GRAPHS * sizeof(unsigned), stream); // key(x) > 0 for all x
    hipMemsetAsync(ssum, 0, NUM_GRAPHS * sizeof(float), stream);
    hipMemsetAsync(pooled, 0, 2 * NUM_GRAPHS * sizeof(float), stream);

    const int nblk = (N_NODES + 255) / 256;
    node_gate_feat<<<nblk, 256, 0, stream>>>(h_cur, gate_w, gate_b, out_w, out_b,
                                             batch, logit, feat, gmax);
    node_exp_sum<<<nblk, 256, 0, stream>>>(logit, batch, gmax, evals, ssum);
    node_pool<<<nblk, 256, 0, stream>>>(evals, feat, batch, ssum, pooled);
    final_softmax<<<1, 64, 0, stream>>>(pooled, out);
}